// gnn_56770877719116
// MI455X (gfx1250) — compile-verified
//
#include <hip/hip_runtime.h>
#include <hip/hip_bf16.h>
#include <math.h>

// ---------------------------------------------------------------------------
// IEGMN GNN forward for MI455X (gfx1250), wave32, WMMA f32 16x16x4.
// B=64, N=512 -> Nt=32768 nodes; E=262144 edges; D_H=128; L=5.
// All GEMMs routed through V_WMMA_F32_16X16X4_F32 (exact f32 semantics).
// ---------------------------------------------------------------------------

#define NT_   32768
#define E_    262144
#define DH_   128
#define L_    5
#define XINIT 0.25f

typedef float v2f __attribute__((ext_vector_type(2)));
typedef float v8f __attribute__((ext_vector_type(8)));

__device__ __forceinline__ v8f wmma4(v2f a, v2f b, v8f c) {
  // (neg_a, A, neg_b, B, c_mod, C, reuse_a, reuse_b)
  return __builtin_amdgcn_wmma_f32_16x16x4_f32(false, a, false, b, (short)0, c,
                                               false, false);
}

__device__ __forceinline__ float siluf(float x) {
  return x / (1.0f + expf(-x));
}

// ------------------------------ utility kernels ----------------------------

__global__ void k_zero(float* p, int n) {
  int i = blockIdx.x * 256 + threadIdx.x;
  if (i < n) p[i] = 0.0f;
}

__global__ void k_copy(float* d, const float* s, int n) {
  int i = blockIdx.x * 256 + threadIdx.x;
  if (i < n) d[i] = s[i];
}

__global__ void k_degree(const int* dst, float* cnt, int n) {
  int i = blockIdx.x * 256 + threadIdx.x;
  if (i < n) atomicAdd(&cnt[dst[i]], 1.0f);
}

__global__ void k_deginv(float* c, int n) {
  int i = blockIdx.x * 256 + threadIdx.x;
  if (i < n) c[i] = 1.0f / fmaxf(c[i], 1.0f);
}

// x = XINIT*x0 + (1-XINIT)*x + agg_x * deg_inv   (elementwise over Nt*3)
__global__ void k_node_x(float* x, const float* coords, const float* aggx,
                         const float* dinv, int n3) {
  int i = blockIdx.x * 256 + threadIdx.x;
  if (i < n3) {
    int node = i / 3;
    x[i] = XINIT * coords[i] + (1.0f - XINIT) * x[i] + aggx[i] * dinv[node];
  }
}

// ------------------------------ embed GEMM ---------------------------------
// h = feat @ W_emb^T ; [Nt,128] x [128,128].  32 nodes / workgroup, 8 waves.
__global__ void __launch_bounds__(256) k_embed(const float* __restrict__ feat,
                                               const float* __restrict__ Wemb,
                                               float* __restrict__ h,
                                               float* __restrict__ h0) {
  __shared__ float in_s[32][132];
  int tid = threadIdx.x;
  int row = tid >> 3, c8 = tid & 7;
  int node = blockIdx.x * 32 + row;
  const float4* sp = (const float4*)(feat + node * 128 + c8 * 16);
  float4* dp = (float4*)&in_s[row][c8 * 16];
  for (int j = 0; j < 4; j++) dp[j] = sp[j];
  __syncthreads();

  int lane = tid & 31, wave = tid >> 5;
  int n0 = wave * 16, ln = lane & 15;
  int koff = (lane >> 4) << 1, mbase = (lane >> 4) << 3;
  v8f acc0, acc1;
  for (int v = 0; v < 8; v++) { acc0[v] = 0.0f; acc1[v] = 0.0f; }
  for (int k0 = 0; k0 < 128; k0 += 4) {
    // B[k][n] = W_emb[n][k]  (transposed access, consecutive k -> b64 load)
    const float* wp = Wemb + (n0 + ln) * 128 + k0 + koff;
    v2f b; b.x = wp[0]; b.y = wp[1];
    v2f a0; a0.x = in_s[ln][k0 + koff];      a0.y = in_s[ln][k0 + koff + 1];
    v2f a1; a1.x = in_s[16 + ln][k0 + koff]; a1.y = in_s[16 + ln][k0 + koff + 1];
    acc0 = wmma4(a0, b, acc0);
    acc1 = wmma4(a1, b, acc1);
  }
  int nb = blockIdx.x * 32;
  for (int v = 0; v < 8; v++) {
    int m = v + mbase;
    int i0 = (nb + m) * 128 + n0 + ln;
    int i1 = (nb + 16 + m) * 128 + n0 + ln;
    h[i0] = acc0[v]; h0[i0] = acc0[v];
    h[i1] = acc1[v]; h0[i1] = acc1[v];
  }
}

// ------------------------------ fused edge kernel --------------------------
// For 32 edges: in = [h[src] | h[dst] | d2]  (16x257 tiles x2)
//   m  = silu(in @ We1 + be1)      (WMMA, K=256 + rank-1 d2 epilogue)
//   m  = silu(m @ We2 + be2)       (WMMA, K=128)
//   coef = tanh(m @ Wx)
//   agg_x[dst] += dx*coef ; agg_m[dst] += m      (global float atomics -> L2)
__global__ void __launch_bounds__(256) k_edge(
    const int* __restrict__ src, const int* __restrict__ dst,
    const float* __restrict__ x, const float* __restrict__ h,
    const float* __restrict__ We1, const float* __restrict__ be1,
    const float* __restrict__ We2, const float* __restrict__ be2,
    const float* __restrict__ Wx, float* __restrict__ aggx,
    float* __restrict__ aggm) {
  __shared__ float in_s[32][260];
  __shared__ float m_s[32][132];
  __shared__ float dx_s[32][3];
  __shared__ float d2_s[32];
  __shared__ int dst_s[32];
  __shared__ float part_s[8][32];

  int tid = threadIdx.x;
  int row = tid >> 3, c8 = tid & 7;
  int e = blockIdx.x * 32 + row;
  int sx = src[e], dd = dst[e];

  const float4* hs = (const float4*)(h + sx * 128 + c8 * 16);
  const float4* hd = (const float4*)(h + dd * 128 + c8 * 16);
  float4* ps = (float4*)&in_s[row][c8 * 16];
  float4* pd = (float4*)&in_s[row][128 + c8 * 16];
  for (int j = 0; j < 4; j++) { ps[j] = hs[j]; pd[j] = hd[j]; }
  if (c8 == 0) {
    float d2 = 0.0f;
    for (int c = 0; c < 3; c++) {
      float dxx = x[sx * 3 + c] - x[dd * 3 + c];
      dx_s[row][c] = dxx;
      d2 += dxx * dxx;
    }
    d2_s[row] = d2;
    in_s[row][256] = d2;
    dst_s[row] = dd;
  }
  __syncthreads();

  int lane = tid & 31, wave = tid >> 5;
  int n0 = wave * 16, ln = lane & 15;
  int koff = (lane >> 4) << 1, mbase = (lane >> 4) << 3;

  // ---- GEMM1: K = 0..255 via WMMA, K=256 (d2) as rank-1 epilogue ----
  float bias1 = be1[n0 + ln];
  v8f acc0, acc1;
  for (int v = 0; v < 8; v++) { acc0[v] = bias1; acc1[v] = bias1; }
  for (int k0 = 0; k0 < 256; k0 += 4) {
    const float* wp = We1 + (k0 + koff) * 128 + n0 + ln;
    v2f b; b.x = wp[0]; b.y = wp[128];
    v2f a0; a0.x = in_s[ln][k0 + koff];      a0.y = in_s[ln][k0 + koff + 1];
    v2f a1; a1.x = in_s[16 + ln][k0 + koff]; a1.y = in_s[16 + ln][k0 + koff + 1];
    acc0 = wmma4(a0, b, acc0);
    acc1 = wmma4(a1, b, acc1);
  }
  float w256 = We1[256 * 128 + n0 + ln];
  for (int v = 0; v < 8; v++) {
    int m = v + mbase;
    acc0[v] = siluf(acc0[v] + d2_s[m] * w256);
    acc1[v] = siluf(acc1[v] + d2_s[16 + m] * w256);
  }
  for (int v = 0; v < 8; v++) {
    int m = v + mbase;
    m_s[m][n0 + ln] = acc0[v];
    m_s[16 + m][n0 + ln] = acc1[v];
  }
  __syncthreads();

  // ---- GEMM2: K = 128 ----
  float bias2 = be2[n0 + ln];
  for (int v = 0; v < 8; v++) { acc0[v] = bias2; acc1[v] = bias2; }
  for (int k0 = 0; k0 < 128; k0 += 4) {
    const float* wp = We2 + (k0 + koff) * 128 + n0 + ln;
    v2f b; b.x = wp[0]; b.y = wp[128];
    v2f a0; a0.x = m_s[ln][k0 + koff];      a0.y = m_s[ln][k0 + koff + 1];
    v2f a1; a1.x = m_s[16 + ln][k0 + koff]; a1.y = m_s[16 + ln][k0 + koff + 1];
    acc0 = wmma4(a0, b, acc0);
    acc1 = wmma4(a1, b, acc1);
  }
  for (int v = 0; v < 8; v++) { acc0[v] = siluf(acc0[v]); acc1[v] = siluf(acc1[v]); }
  __syncthreads();                 // everyone done reading old m_s
  for (int v = 0; v < 8; v++) {
    int m = v + mbase;
    m_s[m][n0 + ln] = acc0[v];
    m_s[16 + m][n0 + ln] = acc1[v];
  }
  __syncthreads();

  // ---- coef = tanh(m @ Wx) ----
  {
    int r2 = tid & 31, ch = tid >> 5;
    const float* wx = Wx + ch * 16;
    float p = 0.0f;
    for (int j = 0; j < 16; j++) p += m_s[r2][ch * 16 + j] * wx[j];
    part_s[ch][r2] = p;
  }
  __syncthreads();
  if (tid < 32) {
    float tot = 0.0f;
    for (int c = 0; c < 8; c++) tot += part_s[c][tid];
    float coef = tanhf(tot);
    int dd2 = dst_s[tid];
    for (int c = 0; c < 3; c++)
      atomicAdd(&aggx[dd2 * 3 + c], dx_s[tid][c] * coef);
  }

  // ---- agg_m[dst] += m ----
  {
    int dd2 = dst_s[row];
    float* dstp = aggm + dd2 * 128 + c8 * 16;
    for (int j = 0; j < 16; j++) atomicAdd(&dstp[j], m_s[row][c8 * 16 + j]);
  }
}

// ------------------------------ node h-update GEMM -------------------------
// c = silu(concat(h, agg_m*deg_inv, h0) @ Wh + bh)   ([Nt,384] x [384,128])
// mode 0: c1 = c        mode 1: h = c - c1
__global__ void __launch_bounds__(256) k_node_h(
    float* __restrict__ h, const float* __restrict__ aggm,
    const float* __restrict__ h0, const float* __restrict__ dinv,
    const float* __restrict__ Wh, const float* __restrict__ bh,
    float* __restrict__ c1, int mode) {
  __shared__ float in_s[32][388];
  int tid = threadIdx.x, row = tid >> 3, c8 = tid & 7;
  int node = blockIdx.x * 32 + row;
  float di = dinv[node];
  const float* hp = h + node * 128 + c8 * 16;
  const float* ap = aggm + node * 128 + c8 * 16;
  const float* op = h0 + node * 128 + c8 * 16;
  for (int j = 0; j < 16; j++) {
    in_s[row][c8 * 16 + j] = hp[j];
    in_s[row][128 + c8 * 16 + j] = ap[j] * di;
    in_s[row][256 + c8 * 16 + j] = op[j];
  }
  __syncthreads();

  int lane = tid & 31, wave = tid >> 5;
  int n0 = wave * 16, ln = lane & 15;
  int koff = (lane >> 4) << 1, mbase = (lane >> 4) << 3;
  float bias = bh[n0 + ln];
  v8f acc0, acc1;
  for (int v = 0; v < 8; v++) { acc0[v] = bias; acc1[v] = bias; }
  for (int k0 = 0; k0 < 384; k0 += 4) {
    const float* wp = Wh + (k0 + koff) * 128 + n0 + ln;
    v2f b; b.x = wp[0]; b.y = wp[128];
    v2f a0; a0.x = in_s[ln][k0 + koff];      a0.y = in_s[ln][k0 + koff + 1];
    v2f a1; a1.x = in_s[16 + ln][k0 + koff]; a1.y = in_s[16 + ln][k0 + koff + 1];
    acc0 = wmma4(a0, b, acc0);
    acc1 = wmma4(a1, b, acc1);
  }
  int nb = blockIdx.x * 32;
  for (int v = 0; v < 8; v++) {
    int m = v + mbase;
    float s0 = siluf(acc0[v]), s1 = siluf(acc1[v]);
    int i0 = (nb + m) * 128 + n0 + ln;
    int i1 = (nb + 16 + m) * 128 + n0 + ln;
    if (mode == 0) { c1[i0] = s0; c1[i1] = s1; }
    else           { h[i0] = s0 - c1[i0]; h[i1] = s1 - c1[i1]; }
  }
}

// ------------------------------ pooling + MLP head -------------------------
__global__ void __launch_bounds__(128) k_head(
    const float* __restrict__ h, const float* __restrict__ cvalid,
    const float* __restrict__ Wfc, const float* __restrict__ bfc,
    const float* __restrict__ Wout, const float* __restrict__ bout,
    float* __restrict__ out) {
  int b = blockIdx.x, d = threadIdx.x;
  __shared__ float z[128];
  __shared__ float red[128];
  float s = 0.0f, cnt = 0.0f;
  for (int n = 0; n < 512; n++) {
    float cv = cvalid[b * 512 + n];
    cnt += cv;
    s += h[(b * 512 + n) * 128 + d] * cv;
  }
  z[d] = s / cnt;
  __syncthreads();
  for (int i = 0; i < 3; i++) {
    float a = bfc[i * 128 + d];
    for (int k = 0; k < 128; k++) a += z[k] * Wfc[i * 128 * 128 + k * 128 + d];
    a = fmaxf(a, 0.0f);
    __syncthreads();
    z[d] = a;
    __syncthreads();
  }
  red[d] = z[d] * Wout[d];
  __syncthreads();
  for (int st = 64; st > 0; st >>= 1) {
    if (d < st) red[d] += red[d + st];
    __syncthreads();
  }
  if (d == 0) out[b] = 1.0f / (1.0f + expf(-(red[0] + bout[0])));
}

// ------------------------------ launch -------------------------------------

extern "C" void kernel_launch(void* const* d_in, const int* in_sizes, int n_in,
                              void* d_out, int out_size, void* d_ws,
                              size_t ws_size, hipStream_t stream) {
  const float* feat   = (const float*)d_in[0];
  const float* coords = (const float*)d_in[1];
  const float* cvalid = (const float*)d_in[2];
  const int* esrc     = (const int*)d_in[3];
  const int* edst     = (const int*)d_in[4];
  const int* csrc     = (const int*)d_in[5];
  const int* cdst     = (const int*)d_in[6];
  const float* Wemb   = (const float*)d_in[7];
  const float* We1    = (const float*)d_in[8];
  const float* be1    = (const float*)d_in[9];
  const float* We2    = (const float*)d_in[10];
  const float* be2    = (const float*)d_in[11];
  const float* Wx     = (const float*)d_in[12];
  const float* Wh     = (const float*)d_in[13];
  const float* bh     = (const float*)d_in[14];
  const float* Wfc    = (const float*)d_in[15];
  const float* bfc    = (const float*)d_in[16];
  const float* Wout   = (const float*)d_in[17];
  const float* bout   = (const float*)d_in[18];
  float* out = (float*)d_out;

  const int Nt = NT_, E = E_;

  // workspace carve (all offsets multiple of 4 floats -> 16B alignment kept)
  float* ws   = (float*)d_ws;
  float* x    = ws;                 // Nt*3
  float* dgi  = x + Nt * 3;         // Nt
  float* dci  = dgi + Nt;           // Nt
  float* aggx = dci + Nt;           // Nt*3
  float* aggm = aggx + Nt * 3;      // Nt*128
  float* h    = aggm + Nt * 128;    // Nt*128
  float* h0   = h + Nt * 128;       // Nt*128
  float* c1   = h0 + Nt * 128;      // Nt*128

  dim3 blk(256);
  int gNt   = (Nt + 255) / 256;
  int gNt3  = (Nt * 3 + 255) / 256;
  int gNtD  = (Nt * 128 + 255) / 256;
  int gE    = (E + 255) / 256;
  int gTile = Nt / 32;   // 1024
  int gEdge = E / 32;    // 8192

  // degree inverses
  k_zero<<<gNt, blk, 0, stream>>>(dgi, Nt);
  k_zero<<<gNt, blk, 0, stream>>>(dci, Nt);
  k_degree<<<gE, blk, 0, stream>>>(edst, dgi, E);
  k_degree<<<gE, blk, 0, stream>>>(cdst, dci, E);
  k_deginv<<<gNt, blk, 0, stream>>>(dgi, Nt);
  k_deginv<<<gNt, blk, 0, stream>>>(dci, Nt);

  // init coords copy + embedding
  k_copy<<<gNt3, blk, 0, stream>>>(x, coords, Nt * 3);
  k_embed<<<gTile, blk, 0, stream>>>(feat, Wemb, h, h0);

  for (int k = 0; k < L_; k++) {
    const float* we1 = We1 + (size_t)k * 257 * 128;
    const float* be1k = be1 + k * 128;
    const float* we2 = We2 + (size_t)k * 128 * 128;
    const float* be2k = be2 + k * 128;
    const float* wxk = Wx + k * 128;
    const float* whk = Wh + (size_t)k * 384 * 128;
    const float* bhk = bh + k * 128;

    // graph sub-layer
    k_zero<<<gNt3, blk, 0, stream>>>(aggx, Nt * 3);
    k_zero<<<gNtD, blk, 0, stream>>>(aggm, Nt * 128);
    k_edge<<<gEdge, blk, 0, stream>>>(esrc, edst, x, h, we1, be1k, we2, be2k,
                                      wxk, aggx, aggm);
    k_node_x<<<gNt3, blk, 0, stream>>>(x, coords, aggx, dgi, Nt * 3);
    k_node_h<<<gTile, blk, 0, stream>>>(h, aggm, h0, dgi, whk, bhk, c1, 0);

    // cross sub-layer (same input h, updated x)
    k_zero<<<gNt3, blk, 0, stream>>>(aggx, Nt * 3);
    k_zero<<<gNtD, blk, 0, stream>>>(aggm, Nt * 128);
    k_edge<<<gEdge, blk, 0, stream>>>(csrc, cdst, x, h, we1, be1k, we2, be2k,
                                      wxk, aggx, aggm);
    k_node_x<<<gNt3, blk, 0, stream>>>(x, coords, aggx, dci, Nt * 3);
    k_node_h<<<gTile, blk, 0, stream>>>(h, aggm, h0, dci, whk, bhk, c1, 1);
  }

  k_head<<<64, dim3(128), 0, stream>>>(h, cvalid, Wfc, bfc, Wout, bout, out);

  (void)in_sizes; (void)n_in; (void)out_size; (void)ws_size;
}